// MacroAUCLossNatural_26371099197875
// MI455X (gfx1250) — compile-verified
//
#include <hip/hip_runtime.h>
#include <hip/hip_bf16.h>

#define B_DIM 512
#define L_DIM 256
#define NT (B_DIM / 16)   // 32 tiles per dimension

typedef __attribute__((ext_vector_type(2))) float v2f;
typedef __attribute__((ext_vector_type(8))) float v8f;

// Fast stable softplus: log(1+exp(d)) = max(d,0) + ln2*log2(1 + exp2(-|d|*log2e))
// Lowers to v_mul(|d|), v_exp_f32, v_add, v_log_f32, v_max, v_fma (~1 ulp trans).
__device__ __forceinline__ float softplus_fast(float d) {
    const float LOG2E = 1.4426950408889634f;
    const float LN2   = 0.6931471805599453f;
    float e = __builtin_amdgcn_exp2f(-fabsf(d) * LOG2E);   // in (0, 1]
    float l = __builtin_amdgcn_logf(1.0f + e);             // log2, arg in (1, 2]
    return fmaxf(d, 0.0f) + LN2 * l;
}

// ---------------------------------------------------------------------------
// Kernel 1: transpose y_pred to [L,B] and build float pos/neg masks [L,B].
// Tiny problem (1.5 MB), entirely L2 resident.
// ---------------------------------------------------------------------------
__global__ void prep_kernel(const int* __restrict__ y_true,
                            const float* __restrict__ y_pred,
                            float* __restrict__ xT,
                            float* __restrict__ pT,
                            float* __restrict__ nT) {
    int idx = blockIdx.x * blockDim.x + threadIdx.x;
    if (idx >= B_DIM * L_DIM) return;
    int b = idx / L_DIM;
    int l = idx % L_DIM;
    float v = y_pred[idx];
    int   t = y_true[idx];
    int   o = l * B_DIM + b;
    xT[o] = v;
    pT[o] = (t == 1) ? 1.0f : 0.0f;
    nT[o] = (t == 0) ? 1.0f : 0.0f;
}

// ---------------------------------------------------------------------------
// Kernel 2: one wave32 per (label l, row tile ti). Per 16x16 (i,j) tile:
//   diff = A_d x B_d  (rank-2: x_j - x_i for all 256 pairs, 1 WMMA)
//   mask = A_m x B_m  (rank-1: pos_i * neg_j,               1 WMMA)
// then 8 per-lane fast softplus + FMA into the C-layout accumulator.
// WMMA (matrix pipe, TRANS-class for co-exec) overlaps the softplus VALU work.
// ---------------------------------------------------------------------------
__global__ void __launch_bounds__(32)
pair_tile_kernel(const float* __restrict__ xT,
                 const float* __restrict__ pT,
                 const float* __restrict__ nT,
                 float* __restrict__ partials) {
    const int l    = blockIdx.y;
    const int ti   = blockIdx.x;
    const int lane = threadIdx.x;
    const int m    = lane & 15;
    const bool lo  = lane < 16;

    const float* xrow = xT + l * B_DIM;
    const float* prow = pT + l * B_DIM;
    const float* nrow = nT + l * B_DIM;

    const float xi = xrow[ti * 16 + m];
    const float pi = prow[ti * 16 + m];

    // A fragment, difference tile: A[m,0]=1, A[m,1]=-x_i[m] (K=2,3 lanes zeroed)
    v2f aD; aD.x = lo ? 1.0f : 0.0f;  aD.y = lo ? -xi : 0.0f;
    // A fragment, mask tile: A[m,0]=pos_i[m]
    v2f aM; aM.x = lo ? pi   : 0.0f;  aM.y = 0.0f;

    v8f acc  = {};
    v8f zero = {};

    #pragma unroll 2
    for (int tj = 0; tj < NT; ++tj) {
        const float xj = xrow[tj * 16 + m];
        const float nj = nrow[tj * 16 + m];
        // B fragments: B_d[0,n]=x_j[n], B_d[1,n]=1 ; B_m[0,n]=neg_j[n]
        v2f bD; bD.x = lo ? xj : 0.0f;  bD.y = lo ? 1.0f : 0.0f;
        v2f bM; bM.x = lo ? nj : 0.0f;  bM.y = 0.0f;

        v8f diff = __builtin_amdgcn_wmma_f32_16x16x4_f32(
            false, aD, false, bD, (short)0, zero, false, false);
        v8f mask = __builtin_amdgcn_wmma_f32_16x16x4_f32(
            false, aM, false, bM, (short)0, zero, false, false);

        #pragma unroll
        for (int v = 0; v < 8; ++v) {
            acc[v] += mask[v] * softplus_fast(diff[v]);
        }
    }

    // 8 VGPRs x 32 lanes cover each of the 256 tile elements exactly once.
    float s = acc[0] + acc[1] + acc[2] + acc[3] + acc[4] + acc[5] + acc[6] + acc[7];
    for (int off = 16; off > 0; off >>= 1)
        s += __shfl_xor(s, off, 32);
    if (lane == 0) partials[l * NT + ti] = s;
}

// ---------------------------------------------------------------------------
// Kernel 3: exact integer pos/neg counts per label, per-label mean, then a
// deterministic shared-memory tree reduction over labels. Single block.
// ---------------------------------------------------------------------------
__global__ void __launch_bounds__(L_DIM)
finalize_kernel(const int* __restrict__ y_true,
                const float* __restrict__ partials,
                float* __restrict__ out) {
    __shared__ float s_tot[L_DIM];
    __shared__ int   s_val[L_DIM];
    const int l = threadIdx.x;

    int pc = 0, nc = 0;
    for (int b = 0; b < B_DIM; ++b) {
        int t = y_true[b * L_DIM + l];
        pc += (t == 1);
        nc += (t == 0);
    }
    float cnt = (float)pc * (float)nc;   // <= 65536, exact in f32

    float sum = 0.0f;
    for (int k = 0; k < NT; ++k) sum += partials[l * NT + k];

    s_tot[l] = (cnt > 0.0f) ? (sum / cnt) : 0.0f;
    s_val[l] = (cnt > 0.0f) ? 1 : 0;
    __syncthreads();

    for (int off = L_DIM / 2; off > 0; off >>= 1) {
        if (l < off) {
            s_tot[l] += s_tot[l + off];
            s_val[l] += s_val[l + off];
        }
        __syncthreads();
    }
    if (l == 0)
        out[0] = (s_val[0] > 0) ? (s_tot[0] / (float)s_val[0]) : 0.0f;
}

// ---------------------------------------------------------------------------
extern "C" void kernel_launch(void* const* d_in, const int* in_sizes, int n_in,
                              void* d_out, int out_size, void* d_ws, size_t ws_size,
                              hipStream_t stream) {
    const int*   y_true = (const int*)d_in[0];
    const float* y_pred = (const float*)d_in[1];

    char* ws = (char*)d_ws;
    const size_t mat = (size_t)B_DIM * L_DIM * sizeof(float);   // 512 KB
    float* xT       = (float*)(ws + 0 * mat);
    float* pT       = (float*)(ws + 1 * mat);
    float* nT       = (float*)(ws + 2 * mat);
    float* partials = (float*)(ws + 3 * mat);                   // 32 KB

    prep_kernel<<<(B_DIM * L_DIM + 255) / 256, 256, 0, stream>>>(
        y_true, y_pred, xT, pT, nT);

    pair_tile_kernel<<<dim3(NT, L_DIM), 32, 0, stream>>>(
        xT, pT, nT, partials);

    finalize_kernel<<<1, L_DIM, 0, stream>>>(
        y_true, partials, (float*)d_out);
}